// PolynomialMorphology_6408091206300
// MI455X (gfx1250) — compile-verified
//
#include <hip/hip_runtime.h>

// PolynomialMorphology (k=3): out = dilation3(erosion3(x)), circular (cross) mask.
// x: (32,1,512,512) fp32.  Fused single pass; TDM async tensor load -> LDS.
// Bandwidth-bound (~67 MB total at 23.3 TB/s ~= 3 us); WMMA inapplicable.

#define IMG_W    512
#define IMG_H    512
#define OUT_ROWS 16                 // output rows per block
#define LOAD_ROWS (OUT_ROWS + 4)    // input strip incl. halo-2 (20)
#define ER_ROWS   (OUT_ROWS + 2)    // erosion strip incl. halo-1 (18)

typedef unsigned int u32x4 __attribute__((ext_vector_type(4)));
typedef int          i32x4 __attribute__((ext_vector_type(4)));
typedef int          i32x8 __attribute__((ext_vector_type(8)));

__global__ __launch_bounds__(256)
void morph_open_fused(const float* __restrict__ x, float* __restrict__ out) {
  __shared__ float xs[LOAD_ROWS][IMG_W];  // 40 KB input strip (global rows y0-2 .. y0+17)
  __shared__ float es[ER_ROWS][IMG_W];    // 36 KB erosion strip (global rows y0-1 .. y0+16)

  const int tid   = threadIdx.x;
  const int strip = blockIdx.x;           // 0 .. IMG_H/OUT_ROWS-1
  const int b     = blockIdx.y;           // image index
  const int y0    = strip * OUT_ROWS;

  const int gr0 = y0 - 2;                             // global row of xs[0]
  int ys = gr0 < 0 ? 0 : gr0;                         // first valid global row
  int ye = y0 + OUT_ROWS + 2; if (ye > IMG_H) ye = IMG_H;
  const int nrows   = ye - ys;                        // valid rows to DMA
  const int ldsrow0 = ys - gr0;                       // 0 or 2

#if __has_builtin(__builtin_amdgcn_tensor_load_to_lds)
  // ---- Tensor Data Mover: DMA nrows x 512 fp32 tile into LDS. Wave 0 only
  // (uniform-per-wave predicate -> s_cbranch_execz skips it on waves 1..7).
  if ((tid >> 5) == 0) {
    unsigned long long gaddr =
        (unsigned long long)(const void*)(x + ((size_t)b * (IMG_H * IMG_W) +
                                               (size_t)ys * IMG_W));
    unsigned int lds_off =
        (unsigned int)(unsigned long long)(const void*)&xs[ldsrow0][0];

    u32x4 g0;                                         // D# group 0 (128b)
    g0[0] = 1u;                                       // count=1, no gather
    g0[1] = lds_off;                                  // lds_addr (bytes)
    g0[2] = (unsigned int)gaddr;                      // global_addr[31:0]
    g0[3] = (unsigned int)((gaddr >> 32) & 0x01FFFFFFu) | (2u << 30); // addr[56:32], type=2

    i32x8 g1;                                         // D# group 1 (256b)
    g1[0] = (int)(2u << 16);                          // workgroup_mask=0, data_size=4B
    g1[1] = (int)((unsigned)IMG_W << 16);             // tensor_dim0[15:0]=512
    g1[2] = (int)((unsigned)nrows << 16);             // tensor_dim1[15:0]=nrows
    g1[3] = (int)((unsigned)IMG_W << 16);             // tile_dim0=512
    g1[4] = nrows;                                    // tile_dim1=nrows, tile_dim2=0
    g1[5] = IMG_W;                                    // tensor_dim0_stride=512
    g1[6] = 0;
    g1[7] = 0;

    i32x4 z4 = {0, 0, 0, 0};                          // groups 2/3 unused (2-D)
#if defined(__clang_major__) && (__clang_major__ >= 23)
    i32x8 z8 = {0, 0, 0, 0, 0, 0, 0, 0};
    __builtin_amdgcn_tensor_load_to_lds(g0, g1, z4, z4, z8, 0);
#else
    __builtin_amdgcn_tensor_load_to_lds(g0, g1, z4, z4, 0);
#endif
  }
#else
  // Fallback: cooperative loads (no TDM builtin on this toolchain).
  for (int idx = tid; idx < nrows * IMG_W; idx += 256) {
    int r = idx >> 9, c = idx & (IMG_W - 1);
    xs[ldsrow0 + r][c] = x[(size_t)b * (IMG_H * IMG_W) + (size_t)(ys + r) * IMG_W + c];
  }
#endif

  // Zero-fill out-of-image halo rows (disjoint from the DMA destination,
  // so no barrier needed between this and the TDM issue).
  for (int r = 0; r < LOAD_ROWS; ++r) {
    int gr = gr0 + r;
    if (gr < 0 || gr >= IMG_H) {
      for (int c = tid; c < IMG_W; c += 256) xs[r][c] = 0.0f;
    }
  }

#if __has_builtin(__builtin_amdgcn_s_wait_tensorcnt)
  __builtin_amdgcn_s_wait_tensorcnt(0);               // TENSORcnt==0 (no-op on waves 1..7)
#endif
  __syncthreads();

  // ---- Stage 1: erosion = clip(prod of 5-tap cross, 0, 1).
  // es row r corresponds to global row y0-1+r; zero-padded xs rows make
  // out-of-image erosion rows come out 0 automatically.
  for (int idx = tid; idx < ER_ROWS * IMG_W; idx += 256) {
    int c = idx & (IMG_W - 1);
    int r = idx >> 9;
    float ctr = xs[r + 1][c];
    float up  = xs[r][c];
    float dn  = xs[r + 2][c];
    float lf  = (c == 0)          ? 0.0f : xs[r + 1][c - 1];
    float rt  = (c == IMG_W - 1)  ? 0.0f : xs[r + 1][c + 1];
    float p = ctr * up * dn * lf * rt;
    es[r][c] = fminf(fmaxf(p, 0.0f), 1.0f);
  }
  __syncthreads();

  // ---- Stage 2: dilation = clip(1 - prod(1 - e_tap), 0, 1), float4 stores.
  // Out-of-range taps have e=0 -> factor (1-0)=1 (identity), via the zeroed
  // es halo rows and the column guards.
  const size_t obase = (size_t)b * (IMG_H * IMG_W) + (size_t)y0 * IMG_W;
  for (int q = tid; q < OUT_ROWS * (IMG_W / 4); q += 256) {
    int row = q >> 7;                  // 128 float4 per row
    int c0  = (q & 127) << 2;
    float4 o;
    float* op = &o.x;
#pragma unroll
    for (int j = 0; j < 4; ++j) {
      int c = c0 + j;
      float ctr = es[row + 1][c];
      float up  = es[row][c];
      float dn  = es[row + 2][c];
      float lf  = (c == 0)         ? 0.0f : es[row + 1][c - 1];
      float rt  = (c == IMG_W - 1) ? 0.0f : es[row + 1][c + 1];
      float p = (1.0f - ctr) * (1.0f - up) * (1.0f - dn) * (1.0f - lf) * (1.0f - rt);
      op[j] = fminf(fmaxf(1.0f - p, 0.0f), 1.0f);
    }
    *(float4*)(out + obase + (size_t)row * IMG_W + c0) = o;
  }
}

extern "C" void kernel_launch(void* const* d_in, const int* in_sizes, int n_in,
                              void* d_out, int out_size, void* d_ws, size_t ws_size,
                              hipStream_t stream) {
  const float* x = (const float*)d_in[0];
  float* out     = (float*)d_out;
  int batch = in_sizes[0] / (IMG_H * IMG_W);          // 32
  dim3 grid(IMG_H / OUT_ROWS, batch, 1);              // 32 strips x 32 images
  morph_open_fused<<<grid, 256, 0, stream>>>(x, out);
}